// QuantumFeatureMap_65481071407826
// MI455X (gfx1250) — compile-verified
//
#include <hip/hip_runtime.h>

// ---------------------------------------------------------------------------
// 20-qubit state-vector sim, 4 layers of (RX·RY per wire) + CNOT chain.
// Wire w <-> state bit (19 - w).
//  * RX,RY fused per wire into one 2x2 complex gate G.
//  * 4 wires grouped -> 16x16 complex unitary, applied as 4 real f32 GEMMs
//    via V_WMMA_F32_16X16X4_F32 (exact f32 math, matrix pipe).
//  * CNOT chain == permutation: out[c] = in[c ^ (c>>1)]  (one gather pass).
// State (8 MB) is L2-resident; pass count is the cost metric.
// ---------------------------------------------------------------------------

typedef float v2f __attribute__((ext_vector_type(2)));
typedef float v8f __attribute__((ext_vector_type(8)));

#define NW      20
#define NSTATES (1u << NW)

struct Cpx { float re, im; };

__device__ __forceinline__ Cpx cmul(Cpx a, Cpx b) {
  Cpx r;
  r.re = fmaf(a.re, b.re, -(a.im * b.im));
  r.im = fmaf(a.re, b.im,  a.im * b.re);
  return r;
}

// |0...0> init
__global__ void init_state_k(float2* st) {
  unsigned t = blockIdx.x * blockDim.x + threadIdx.x;
  st[t] = (t == 0u) ? make_float2(1.f, 0.f) : make_float2(0.f, 0.f);
}

__global__ void zero_out_k(float* out) {
  if (threadIdx.x < NW) out[threadIdx.x] = 0.f;
}

// ---------------------------------------------------------------------------
// Apply U = G(b+3) (x) G(b+2) (x) G(b+1) (x) G(b)  (16x16 complex) to the
// state bits [b, b+4). One wave handles a 16(row m) x 16(column f) tile:
//   Dre = Ur*Sre - Ui*Sim ; Dim = Ui*Sre + Ur*Sim     (4 real 16x16x16 GEMMs)
// Each real GEMM = 4 chained V_WMMA_F32_16X16X4_F32 (K accumulated 4 at a time).
// Tiles are disjoint -> safe in place. EXEC is all-ones (no divergence).
// grid 512 x block 256 (8 waves)  => 4096 tiles * 16 cols = 2^16 columns.
// ---------------------------------------------------------------------------
__global__ __launch_bounds__(256) void group_u4_wmma_k(float2* st,
                                                       const float* __restrict__ x,
                                                       int b) {
  const int lane = threadIdx.x & 31;
  const int half = lane >> 4;   // K/M half select per ISA fragment layout
  const int r    = lane & 15;   // A-matrix row
  const int n    = lane & 15;   // B/D column within tile
  const unsigned wtile = blockIdx.x * 8u + (unsigned)(threadIdx.x >> 5);

  // Fused RX*RY gate per wire; bit (b+j) <-> wire (19-b-j).
  Cpx g[4][2][2];
#pragma unroll
  for (int j = 0; j < 4; ++j) {
    float th = 0.5f * x[19 - b - j];
    float c = cosf(th), s = sinf(th);
    float cc = c * c, ss = s * s, cs = c * s;
    g[j][0][0].re = cc;  g[j][0][0].im = ss;    // c^2 + i s^2
    g[j][0][1].re = -cs; g[j][0][1].im = -cs;   // -cs - i cs
    g[j][1][0].re = cs;  g[j][1][0].im = -cs;   //  cs - i cs
    g[j][1][1].re = cc;  g[j][1][1].im = -ss;   // c^2 - i s^2
  }

  auto uent = [&](int m, int k) -> Cpx {
    Cpx u = cmul(g[0][m & 1][k & 1],        g[1][(m >> 1) & 1][(k >> 1) & 1]);
    Cpx v = cmul(g[2][(m >> 2) & 1][(k >> 2) & 1], g[3][(m >> 3) & 1][(k >> 3) & 1]);
    return cmul(u, v);
  };

  // A fragments (16x4 f32 per chunk): lane<16 -> K = k0,k0+1 ; lane>=16 -> K = k0+2,k0+3
  v2f aRe[4], aIm[4], aImN[4];
#pragma unroll
  for (int c = 0; c < 4; ++c) {
    int k0 = c * 4 + half * 2;
    Cpx e0 = uent(r, k0);
    Cpx e1 = uent(r, k0 + 1);
    aRe[c].x  = e0.re;  aRe[c].y  = e1.re;
    aIm[c].x  = e0.im;  aIm[c].y  = e1.im;
    aImN[c].x = -e0.im; aImN[c].y = -e1.im;   // f32 WMMA has no A-neg modifier
  }

  // Column f -> state index: idx(k, f) = (hi << (b+4)) | (k << b) | lo
  const unsigned f    = wtile * 16u + (unsigned)n;
  const unsigned lo   = f & ((1u << b) - 1u);
  const unsigned hi   = f >> b;
  const unsigned base = (hi << (b + 4)) | lo;

  // B fragments (4x16 f32 per chunk): same half-split of K as A, N = lane&15.
  v2f bRe[4], bIm[4];
#pragma unroll
  for (int c = 0; c < 4; ++c) {
    int k0 = c * 4 + half * 2;
    float2 s0 = st[base + ((unsigned)k0 << b)];
    float2 s1 = st[base + ((unsigned)(k0 + 1) << b)];
    bRe[c].x = s0.x; bRe[c].y = s1.x;
    bIm[c].x = s0.y; bIm[c].y = s1.y;
  }

  v8f dRe = {0, 0, 0, 0, 0, 0, 0, 0};
  v8f dIm = {0, 0, 0, 0, 0, 0, 0, 0};
#pragma unroll
  for (int c = 0; c < 4; ++c)
    dRe = __builtin_amdgcn_wmma_f32_16x16x4_f32(false, aRe[c],  false, bRe[c], (short)0, dRe, false, false);
#pragma unroll
  for (int c = 0; c < 4; ++c)
    dRe = __builtin_amdgcn_wmma_f32_16x16x4_f32(false, aImN[c], false, bIm[c], (short)0, dRe, false, false);
#pragma unroll
  for (int c = 0; c < 4; ++c)
    dIm = __builtin_amdgcn_wmma_f32_16x16x4_f32(false, aIm[c],  false, bRe[c], (short)0, dIm, false, false);
#pragma unroll
  for (int c = 0; c < 4; ++c)
    dIm = __builtin_amdgcn_wmma_f32_16x16x4_f32(false, aRe[c],  false, bIm[c], (short)0, dIm, false, false);

  // D layout: value (M = v + 8*half, N = lane&15) in accumulator element v.
#pragma unroll
  for (int v = 0; v < 8; ++v) {
    unsigned m = (unsigned)(v + half * 8);
    st[base + (m << b)] = make_float2(dRe[v], dIm[v]);
  }
}

// CNOT chain CNOT(0,1)..CNOT(18,19): new bit j = XOR of old bits >= j
// (wire w is bit 19-w), inverse old = c ^ (c>>1). Coalesced writes.
__global__ void perm_k(const float2* __restrict__ in, float2* __restrict__ out) {
  unsigned t = blockIdx.x * blockDim.x + threadIdx.x;   // 2^18 threads
#pragma unroll
  for (int r = 0; r < 4; ++r) {
    unsigned c = t + ((unsigned)r << 18);
    out[c] = in[c ^ (c >> 1)];
  }
}

// <Z_i> = sum_s (-1)^{bit(19-i) of s} * |amp_s|^2
__global__ void expz_k(const float2* __restrict__ st, float* out) {
  unsigned t = blockIdx.x * blockDim.x + threadIdx.x;   // 65536 threads
  float acc[NW];
#pragma unroll
  for (int i = 0; i < NW; ++i) acc[i] = 0.f;

#pragma unroll
  for (int r = 0; r < 16; ++r) {
    unsigned idx = t + ((unsigned)r << 16);
    float2 a = st[idx];
    float p = fmaf(a.x, a.x, a.y * a.y);
#pragma unroll
    for (int i = 0; i < NW; ++i) {
      float sgn = 1.f - 2.f * (float)((idx >> (19 - i)) & 1u);
      acc[i] = fmaf(sgn, p, acc[i]);
    }
  }
  // wave32 reduction
#pragma unroll
  for (int off = 16; off > 0; off >>= 1)
#pragma unroll
    for (int i = 0; i < NW; ++i)
      acc[i] += __shfl_down(acc[i], off, 32);

  if ((threadIdx.x & 31) == 0)
#pragma unroll
    for (int i = 0; i < NW; ++i)
      atomicAdd(&out[i], acc[i]);
}

extern "C" void kernel_launch(void* const* d_in, const int* in_sizes, int n_in,
                              void* d_out, int out_size, void* d_ws, size_t ws_size,
                              hipStream_t stream) {
  (void)in_sizes; (void)n_in; (void)out_size; (void)ws_size;
  const float* x = (const float*)d_in[0];
  float* out = (float*)d_out;

  float2* A = (float2*)d_ws;          // 8 MB
  float2* B = A + NSTATES;            // 8 MB (ping-pong for permutation)

  zero_out_k<<<1, 32, 0, stream>>>(out);
  init_state_k<<<NSTATES / 256, 256, 0, stream>>>(A);

  float2* cur = A;
  float2* oth = B;
  for (int layer = 0; layer < 4; ++layer) {
    for (int g = 0; g < 5; ++g)                      // bits [4g, 4g+4), commute
      group_u4_wmma_k<<<512, 256, 0, stream>>>(cur, x, 4 * g);
    perm_k<<<1024, 256, 0, stream>>>(cur, oth);      // fused CNOT chain
    float2* tmp = cur; cur = oth; oth = tmp;
  }

  expz_k<<<256, 256, 0, stream>>>(cur, out);
}